// ReviewCredibilityGAT_76149770158551
// MI455X (gfx1250) — compile-verified
//
#include <hip/hip_runtime.h>
#include <math.h>

// ---------------------------------------------------------------------------
// ReviewCredibilityGAT: q/k projections (WMMA f32 GEMM) + segment softmax.
// MI455X (gfx1250), wave32. Matrix math via V_WMMA_F32_16X16X4_F32.
// ---------------------------------------------------------------------------

typedef float v2f __attribute__((ext_vector_type(2)));
typedef float v8f __attribute__((ext_vector_type(8)));

#define IN_DIM  256
#define HIDDEN  32
#define NSEG    100000     // matches reference num_segments
#define WPITCH  260        // LDS row pitch (floats): 4*n mod 64 distinct -> conflict-free b64

// Monotone float -> uint key so unsigned atomicMax == float max.
__device__ __forceinline__ unsigned fkey(float f) {
  unsigned u = __float_as_uint(f);
  return (u & 0x80000000u) ? ~u : (u | 0x80000000u);
}
__device__ __forceinline__ float funkey(unsigned k) {
  return (k & 0x80000000u) ? __uint_as_float(k & 0x7FFFFFFFu)
                           : __uint_as_float(~k);
}

// ---------------------------------------------------------------------------
__global__ void seg_init_kernel(unsigned* __restrict__ seg_key,
                                float* __restrict__ seg_sum, int nseg) {
  int i = blockIdx.x * blockDim.x + threadIdx.x;
  if (i < nseg) { seg_key[i] = 0u; seg_sum[i] = 0.0f; }
}

// ---------------------------------------------------------------------------
// Main GEMM + attention-logit kernel.
// Wave = 64 rows (4 M-tiles). N-tiles: t=0,1 -> q cols 0..31 ; t=2,3 -> k cols.
// A frag (16x4 f32): lane L holds M=L%16, K = v + 2*(L/16)  (v = vgpr 0/1)
// B frag (4x16 f32): lane L holds N=L%16, K = v + 2*(L/16)  (mirrors A per ISA 7.12.2)
// C/D  (16x16 f32): vgpr e, lanes 0-15 -> M=e ; lanes 16-31 -> M=e+8 ; N=L%16
// ---------------------------------------------------------------------------
__global__ __launch_bounds__(256) void att_wmma_kernel(
    const float* __restrict__ x,
    const float* __restrict__ Wq, const float* __restrict__ bq,
    const float* __restrict__ Wk, const float* __restrict__ bk,
    const int*   __restrict__ idx,
    float*       __restrict__ att_out,      // d_out, holds att temporarily
    unsigned*    __restrict__ seg_key,
    int n)
{
  extern __shared__ float lds_w[];   // 64 rows x WPITCH floats (W^T, q then k)

  // Stage [Wq | Wk]^T into LDS: row h = hidden index (0..63), col = kk (0..255)
  for (int i = threadIdx.x; i < IN_DIM * HIDDEN; i += blockDim.x) {
    int kk = i >> 5;           // input-dim index
    int nn = i & 31;           // hidden index
    lds_w[nn * WPITCH + kk]        = Wq[i];
    lds_w[(32 + nn) * WPITCH + kk] = Wk[i];
  }
  __syncthreads();

  const int lane = threadIdx.x & 31;
  const int lq   = lane & 15;     // N / M position within tile
  const int half = lane >> 4;     // K-pair selector
  const int wid  = blockIdx.x * (blockDim.x >> 5) + (threadIdx.x >> 5);
  const long long row_base = (long long)wid * 64;
  if (row_base >= n) return;      // whole-wave uniform: EXEC stays all-1s

  // Per-M-tile global row pointers (clamped; only affects discarded tails)
  const float* ap[4];
#pragma unroll
  for (int r = 0; r < 4; ++r) {
    long long row = row_base + r * 16 + lq;
    if (row > (long long)n - 1) row = (long long)n - 1;
    ap[r] = x + row * IN_DIM + 2 * half;
  }
  // Per-N-tile LDS element offsets for B fragments
  int bo[4];
#pragma unroll
  for (int t = 0; t < 4; ++t)
    bo[t] = (t * 16 + lq) * WPITCH + 2 * half;

  v8f acc[4][4];
#pragma unroll
  for (int r = 0; r < 4; ++r)
#pragma unroll
    for (int t = 0; t < 4; ++t) {
      v8f z = {0.f, 0.f, 0.f, 0.f, 0.f, 0.f, 0.f, 0.f};
      acc[r][t] = z;
    }

  // K loop: 64 steps of K=4. 4 global b64 + 4 ds b64 loads feed 16 WMMAs.
#pragma unroll 2
  for (int k = 0; k < IN_DIM; k += 4) {
    v2f a[4], b[4];
#pragma unroll
    for (int r = 0; r < 4; ++r)
      a[r] = *(const v2f*)(ap[r] + k);
#pragma unroll
    for (int t = 0; t < 4; ++t)
      b[t] = *(const v2f*)(&lds_w[bo[t] + k]);
#pragma unroll
    for (int r = 0; r < 4; ++r)
#pragma unroll
      for (int t = 0; t < 4; ++t)
        acc[r][t] = __builtin_amdgcn_wmma_f32_16x16x4_f32(
            /*neg_a=*/false, a[r], /*neg_b=*/false, b[t],
            /*c_mod=*/(short)0, acc[r][t],
            /*reuse_a=*/false, /*reuse_b=*/false);
  }

  // Epilogue: att = (q + bq) . (k + bk) per row, reduce across N within
  // each 16-lane half (C/D layout: N = lane%16).
  const float bq0 = bq[lq], bq1 = bq[lq + 16];
  const float bk0 = bk[lq], bk1 = bk[lq + 16];

#pragma unroll
  for (int r = 0; r < 4; ++r) {
#pragma unroll
    for (int e = 0; e < 8; ++e) {
      float q0 = acc[r][0][e] + bq0;
      float q1 = acc[r][1][e] + bq1;
      float k0 = acc[r][2][e] + bk0;
      float k1 = acc[r][3][e] + bk1;
      float p = q0 * k0 + q1 * k1;          // 2 of 32 hidden terms per lane
      p += __shfl_xor(p, 1, 32);            // masks 1..8 stay within each
      p += __shfl_xor(p, 2, 32);            // 16-lane half (wave32)
      p += __shfl_xor(p, 4, 32);
      p += __shfl_xor(p, 8, 32);
      long long row = row_base + r * 16 + e + 8 * half;
      if (lq == 0 && row < n) {
        att_out[row] = p;
        atomicMax(&seg_key[idx[row]], fkey(p));
      }
    }
  }
}

// ---------------------------------------------------------------------------
__global__ void seg_expsum_kernel(float* __restrict__ att_e,   // in: att, out: e
                                  const int* __restrict__ idx,
                                  const unsigned* __restrict__ seg_key,
                                  float* __restrict__ seg_sum, int n) {
  int i = blockIdx.x * blockDim.x + threadIdx.x;
  if (i < n) {
    int s = idx[i];
    float m = funkey(seg_key[s]);
    if (!__builtin_isfinite(m)) m = 0.0f;   // mirror reference safety
    float e = __expf(att_e[i] - m);
    att_e[i] = e;
    atomicAdd(&seg_sum[s], e);
  }
}

__global__ void seg_norm_kernel(float* __restrict__ w,
                                const int* __restrict__ idx,
                                const float* __restrict__ seg_sum, int n) {
  int i = blockIdx.x * blockDim.x + threadIdx.x;
  if (i < n) w[i] = w[i] / seg_sum[idx[i]];
}

// ---------------------------------------------------------------------------
extern "C" void kernel_launch(void* const* d_in, const int* in_sizes, int n_in,
                              void* d_out, int out_size, void* d_ws, size_t ws_size,
                              hipStream_t stream) {
  const float* x   = (const float*)d_in[0];
  const float* Wq  = (const float*)d_in[1];
  const float* bq  = (const float*)d_in[2];
  const float* Wk  = (const float*)d_in[3];
  const float* bk  = (const float*)d_in[4];
  const int*   idx = (const int*)d_in[5];
  const int n = in_sizes[5];                // N rows

  float*    out     = (float*)d_out;
  unsigned* seg_key = (unsigned*)d_ws;
  float*    seg_sum = (float*)((char*)d_ws + sizeof(unsigned) * (size_t)NSEG);

  // 1) reset segment workspace (ws is not re-poisoned between replays)
  seg_init_kernel<<<(NSEG + 255) / 256, 256, 0, stream>>>(seg_key, seg_sum, NSEG);

  // 2) WMMA GEMM + att + segment max (64 rows per wave, 8 waves per block)
  const int waves  = (n + 63) / 64;
  const int blocks = (waves + 7) / 8;
  const size_t lds_bytes = (size_t)64 * WPITCH * sizeof(float);
  att_wmma_kernel<<<blocks, 256, lds_bytes, stream>>>(
      x, Wq, bq, Wk, bk, idx, out, seg_key, n);

  // 3) exp + segment sum
  seg_expsum_kernel<<<(n + 255) / 256, 256, 0, stream>>>(out, idx, seg_key, seg_sum, n);

  // 4) normalize
  seg_norm_kernel<<<(n + 255) / 256, 256, 0, stream>>>(out, idx, seg_sum, n);
}